// EfficientRelativeMultiheadAttention_31903017074725
// MI455X (gfx1250) — compile-verified
//
#include <hip/hip_runtime.h>

// ---------------------------------------------------------------------------
// EfficientRelativeMultiheadAttention for MI455X (gfx1250, wave32, WMMA f32,
// Tensor Data Mover for K/V tile staging)
//
//   B=2, S=2048, E=512, H=8, Dk=Dv=64, RELATIVE_CUTOFF=128
//
// Pipeline (all intermediates live in d_ws; everything fits in 192MB L2):
//   K1: Q,K,V = xs @ w_{q,k,v}          (f32 WMMA 16x16x4, A-reuse x3)
//   K2: P     = Q @ a_k^T  (S x 272pad) (bias table; skew == gather on P)
//   K3: fused flash attention; K/V tiles DMA'd to LDS by the TDM
//       (tensor_load_to_lds, double-buffered, TENSORcnt-gated)
//   K4: out[b,s,v] = sum_h w_o[h] * Z[b,h,s,v]   (deterministic reduce)
// ---------------------------------------------------------------------------

typedef __attribute__((ext_vector_type(2))) float v2f;
typedef __attribute__((ext_vector_type(8))) float v8f;
typedef unsigned int u32x4 __attribute__((ext_vector_type(4)));
typedef int i32x4 __attribute__((ext_vector_type(4)));
typedef int i32x8 __attribute__((ext_vector_type(8)));

#define WMMA_F32(a, b, c) \
  __builtin_amdgcn_wmma_f32_16x16x4_f32(false, (a), false, (b), (short)0, (c), false, false)

constexpr int B_ = 2, S_ = 2048, E_ = 512, H_ = 8, D_ = 64;
constexpr int REL_ = 128;      // RELATIVE_CUTOFF
constexpr int PW_  = 272;      // P row width: 257 padded to multiple of 16
constexpr int JT_  = 32;       // keys per TDM tile in the attention kernel

// ---------------------------------------------------------------------------
// TDM: DMA a (rows x 64) f32 row-major tile from global memory into LDS.
// D# built per CDNA5 ISA (group0/group1; 2D tile, so
// groups 2/3 are zero). Tracked with TENSORcnt. EXEC is ignored; issued by
// one wave. This toolchain exposes the 6-arg builtin form.
// ---------------------------------------------------------------------------
__device__ __forceinline__ void tdm_load_tile_f32(uint32_t lds_off_bytes,
                                                  const float* gptr,
                                                  int rows) {
  const unsigned long long ga = (unsigned long long)(uintptr_t)gptr;
  u32x4 g0;
  g0.x = 1u;                                        // count=1 (valid descriptor)
  g0.y = lds_off_bytes;                             // lds_addr
  g0.z = (uint32_t)ga;                              // global_addr[31:0]
  g0.w = ((uint32_t)(ga >> 32) & 0x01FFFFFFu)       // global_addr[56:32]
         | 0x80000000u;                             // type=2 ("image")
  i32x8 g1;
  g1[0] = 0x20000;                                  // data_size=2 (4 bytes)
  g1[1] = (D_ << 16);                               // tensor_dim0 = 64
  g1[2] = (rows << 16);                             // tensor_dim1[15:0] = rows
  g1[3] = (D_ << 16);                               // tile_dim0 = 64 (dim1 hi=0)
  g1[4] = rows;                                     // tile_dim1 = rows
  g1[5] = D_;                                       // tensor_dim0_stride = 64
  g1[6] = 0;
  g1[7] = 0;
  const i32x4 z4 = {0, 0, 0, 0};
  const i32x8 z8 = {0, 0, 0, 0, 0, 0, 0, 0};
  __builtin_amdgcn_tensor_load_to_lds(g0, g1, z4, z4, z8, 0);
}

// ---------------------------------------------------------------------------
// Kernel 1: Q/K/V projection. Block = 256 thr = 8 waves; wave -> 16 rows.
// Grid (S/128, H, B). One A fragment feeds 12 WMMAs (Q,K,V x 4 n-tiles).
// ---------------------------------------------------------------------------
__global__ __launch_bounds__(256) void qkv_proj_kernel(
    const float* __restrict__ xs, const float* __restrict__ wq,
    const float* __restrict__ wk, const float* __restrict__ wv,
    float* __restrict__ Q, float* __restrict__ K, float* __restrict__ V) {
  const int wave = threadIdx.x >> 5;
  const int lane = threadIdx.x & 31;
  const int g = lane >> 4;          // half-wave group (selects K pair)
  const int n = lane & 15;          // row (A) / col (B,C)
  const int h = blockIdx.y, b = blockIdx.z;
  const int m0 = blockIdx.x * 128 + wave * 16;
  const int bh = b * H_ + h;

  const float* xrow = xs + (size_t)(b * S_ + m0 + n) * E_;
  const float* wqh = wq + (size_t)h * E_ * D_;
  const float* wkh = wk + (size_t)h * E_ * D_;
  const float* wvh = wv + (size_t)h * E_ * D_;

  v8f cq[4], ck[4], cv[4];
#pragma unroll
  for (int t = 0; t < 4; ++t) {
    cq[t] = (v8f){0.f, 0.f, 0.f, 0.f, 0.f, 0.f, 0.f, 0.f};
    ck[t] = cq[t];
    cv[t] = cq[t];
  }

  for (int kk = 0; kk < E_; kk += 4) {
    v2f a = *(const v2f*)(xrow + kk + 2 * g);          // A 16x4 fragment
    const int r0 = (kk + 2 * g) * D_ + n;              // B row (k = kk+2g)
    const int r1 = r0 + D_;                            // B row (k = kk+2g+1)
#pragma unroll
    for (int t = 0; t < 4; ++t) {
      const int o = t * 16;
      v2f bq; bq.x = wqh[r0 + o]; bq.y = wqh[r1 + o];
      v2f bk; bk.x = wkh[r0 + o]; bk.y = wkh[r1 + o];
      v2f bv; bv.x = wvh[r0 + o]; bv.y = wvh[r1 + o];
      cq[t] = WMMA_F32(a, bq, cq[t]);
      ck[t] = WMMA_F32(a, bk, ck[t]);
      cv[t] = WMMA_F32(a, bv, cv[t]);
    }
  }
#pragma unroll
  for (int t = 0; t < 4; ++t)
#pragma unroll
    for (int v = 0; v < 8; ++v) {
      const size_t o = ((size_t)bh * S_ + m0 + v + 8 * g) * D_ + t * 16 + n;
      Q[o] = cq[t][v];
      K[o] = ck[t][v];
      V[o] = cv[t][v];
    }
}

// ---------------------------------------------------------------------------
// Kernel 2: P = Q @ a_k^T, output (bh, S, 272) with cols >= 257 zero.
// Grid (S/128, H, B), 8 waves x 16 rows, Q fragments resident.
// ---------------------------------------------------------------------------
__global__ __launch_bounds__(256) void qrel_kernel(
    const float* __restrict__ Q, const float* __restrict__ ak,
    float* __restrict__ P) {
  const int wave = threadIdx.x >> 5;
  const int lane = threadIdx.x & 31;
  const int g = lane >> 4, n = lane & 15;
  const int h = blockIdx.y, b = blockIdx.z;
  const int i0 = blockIdx.x * 128 + wave * 16;
  const int bh = b * H_ + h;

  const float* qrow = Q + ((size_t)bh * S_ + i0 + n) * D_;
  v2f aq[16];
#pragma unroll
  for (int kk = 0; kk < 16; ++kk) aq[kk] = *(const v2f*)(qrow + kk * 4 + 2 * g);

  for (int nt = 0; nt < PW_ / 16; ++nt) {
    v8f c = (v8f){0.f, 0.f, 0.f, 0.f, 0.f, 0.f, 0.f, 0.f};
    const int row = nt * 16 + n;          // a_k row index (0..271)
    const bool valid = row < 2 * REL_ + 1;
#pragma unroll
    for (int kk = 0; kk < 16; ++kk) {
      v2f bb;
      if (valid) {
        bb = *(const v2f*)(ak + (size_t)row * D_ + kk * 4 + 2 * g);
      } else {
        bb.x = 0.f; bb.y = 0.f;
      }
      c = WMMA_F32(aq[kk], bb, c);
    }
#pragma unroll
    for (int v = 0; v < 8; ++v)
      P[(size_t)bh * S_ * PW_ + (size_t)(i0 + v + 8 * g) * PW_ + nt * 16 + n] = c[v];
  }
}

// ---------------------------------------------------------------------------
// Kernel 3: fused flash attention. Block = 128 thr = 4 waves, wave -> 16 q
// rows (block covers 64). Grid (S/64, H, B). K/V arrive in LDS via TDM,
// double-buffered in JT_=32-key tiles: wave 0 issues tile jt+1 right after
// the barrier that publishes tile jt, so the DMA overlaps all compute.
// Per 16-key sub-tile: 16 WMMA (QK^T) + online softmax (shfl reductions in
// the 16-lane half owning each row) + LDS transpose of the prob tile to
// A-layout + 16 WMMA (P@V).
// ---------------------------------------------------------------------------
__global__ __launch_bounds__(128) void attn_kernel(
    const float* __restrict__ Q, const float* __restrict__ K,
    const float* __restrict__ V, const float* __restrict__ P,
    float* __restrict__ Z) {
  __shared__ float kbuf[2][JT_ * D_];   // 2 x 8 KB
  __shared__ float vbuf[2][JT_ * D_];   // 2 x 8 KB
  __shared__ float xch[4][16][17];      // wave-private prob tiles (padded)

  const int wave = threadIdx.x >> 5;
  const int lane = threadIdx.x & 31;
  const int g = lane >> 4, n = lane & 15;
  const int h = blockIdx.y, b = blockIdx.z;
  const int bh = b * H_ + h;
  const int i0 = (blockIdx.x * 4 + wave) * 16;

  const float* Kb = K + (size_t)bh * S_ * D_;
  const float* Vb = V + (size_t)bh * S_ * D_;
  const float* Pb = P + (size_t)bh * S_ * PW_;
  const float* qrow = Q + ((size_t)bh * S_ + i0 + n) * D_;

  v2f aq[16];
#pragma unroll
  for (int kk = 0; kk < 16; ++kk) aq[kk] = *(const v2f*)(qrow + kk * 4 + 2 * g);

  float Mx[8], L[8];
  v8f acc[4];
#pragma unroll
  for (int v = 0; v < 8; ++v) { Mx[v] = -1e30f; L[v] = 0.f; }
#pragma unroll
  for (int t = 0; t < 4; ++t) acc[t] = (v8f){0.f, 0.f, 0.f, 0.f, 0.f, 0.f, 0.f, 0.f};

  const uint32_t kof[2] = {(uint32_t)(uintptr_t)&kbuf[0][0],
                           (uint32_t)(uintptr_t)&kbuf[1][0]};
  const uint32_t vof[2] = {(uint32_t)(uintptr_t)&vbuf[0][0],
                           (uint32_t)(uintptr_t)&vbuf[1][0]};

  // prologue: DMA tile 0
  if (wave == 0) {
    tdm_load_tile_f32(kof[0], Kb, JT_);
    tdm_load_tile_f32(vof[0], Vb, JT_);
  }

  for (int jt = 0; jt < S_ / JT_; ++jt) {
    if (wave == 0) __builtin_amdgcn_s_wait_tensorcnt(0);  // tile jt landed
    __syncthreads();                                      // publish tile jt
    if (wave == 0 && jt + 1 < S_ / JT_) {                 // prefetch tile jt+1
      tdm_load_tile_f32(kof[(jt + 1) & 1], Kb + (size_t)(jt + 1) * JT_ * D_, JT_);
      tdm_load_tile_f32(vof[(jt + 1) & 1], Vb + (size_t)(jt + 1) * JT_ * D_, JT_);
    }
    const float* kt = kbuf[jt & 1];
    const float* vt = vbuf[jt & 1];

#pragma unroll
    for (int sub = 0; sub < 2; ++sub) {
      const int j0 = jt * JT_ + sub * 16;   // global key index of sub-tile
      const int jl = sub * 16;              // local row within LDS tile

      // ---- S_tile = Q(16x64) @ K^T(64x16), K fragments from LDS ----
      v8f cs = (v8f){0.f, 0.f, 0.f, 0.f, 0.f, 0.f, 0.f, 0.f};
#pragma unroll
      for (int kk = 0; kk < 16; ++kk) {
        v2f bb = *(const v2f*)(kt + (jl + n) * D_ + kk * 4 + 2 * g);
        cs = WMMA_F32(aq[kk], bb, cs);
      }

      // ---- bias + online softmax (row m = v + 8g, col = n) ----
      float pr[8], sc[8];
#pragma unroll
      for (int v = 0; v < 8; ++v) {
        const int m = v + 8 * g;
        int d = j0 + n - (i0 + m);
        d = min(max(d, -REL_), REL_) + REL_;
        float e = (cs[v] + Pb[(size_t)(i0 + m) * PW_ + d]) * 0.125f;
        float mv = e;
        mv = fmaxf(mv, __shfl_xor(mv, 1, 16));
        mv = fmaxf(mv, __shfl_xor(mv, 2, 16));
        mv = fmaxf(mv, __shfl_xor(mv, 4, 16));
        mv = fmaxf(mv, __shfl_xor(mv, 8, 16));
        const float Mn = fmaxf(Mx[v], mv);
        const float s = __expf(Mx[v] - Mn);
        const float p = __expf(e - Mn);
        float rs = p;
        rs += __shfl_xor(rs, 1, 16);
        rs += __shfl_xor(rs, 2, 16);
        rs += __shfl_xor(rs, 4, 16);
        rs += __shfl_xor(rs, 8, 16);
        L[v] = L[v] * s + rs;
        Mx[v] = Mn;
        pr[v] = p;
        sc[v] = s;
      }
#pragma unroll
      for (int t = 0; t < 4; ++t)
#pragma unroll
        for (int v = 0; v < 8; ++v) acc[t][v] = acc[t][v] * sc[v];

      // ---- D-layout -> A-layout transpose of prob tile (wave-private LDS,
      //      LDS ops are in-order per wave) ----
#pragma unroll
      for (int v = 0; v < 8; ++v) xch[wave][v + 8 * g][n] = pr[v];
      asm volatile("s_wait_dscnt 0" ::: "memory");
      v2f ap[4];
#pragma unroll
      for (int kkp = 0; kkp < 4; ++kkp) {
        ap[kkp].x = xch[wave][n][kkp * 4 + 2 * g];
        ap[kkp].y = xch[wave][n][kkp * 4 + 2 * g + 1];
      }

      // ---- acc += P(16x16) @ V(16x64), V fragments from LDS ----
#pragma unroll
      for (int t = 0; t < 4; ++t)
#pragma unroll
        for (int kkp = 0; kkp < 4; ++kkp) {
          const int r = jl + kkp * 4 + 2 * g;
          v2f bv;
          bv.x = vt[r * D_ + t * 16 + n];
          bv.y = vt[(r + 1) * D_ + t * 16 + n];
          acc[t] = WMMA_F32(ap[kkp], bv, acc[t]);
        }
    }
  }

  // ---- normalize and store Z ----
  float inv[8];
#pragma unroll
  for (int v = 0; v < 8; ++v) inv[v] = 1.f / L[v];
#pragma unroll
  for (int t = 0; t < 4; ++t)
#pragma unroll
    for (int v = 0; v < 8; ++v)
      Z[((size_t)bh * S_ + i0 + v + 8 * g) * D_ + t * 16 + n] = acc[t][v] * inv[v];
}

// ---------------------------------------------------------------------------
// Kernel 4: out[b,s,v] = sum_h w_o[h] * Z[b,h,s,v]  (deterministic)
// ---------------------------------------------------------------------------
__global__ __launch_bounds__(256) void head_reduce_kernel(
    const float* __restrict__ Z, const float* __restrict__ wo,
    float* __restrict__ out) {
  const int idx = blockIdx.x * 256 + threadIdx.x;
  const int b = idx / (S_ * D_);
  const int sv = idx - b * (S_ * D_);
  float a = 0.f;
#pragma unroll
  for (int h = 0; h < H_; ++h)
    a += wo[h] * Z[(size_t)(b * H_ + h) * S_ * D_ + sv];
  out[idx] = a;
}

// ---------------------------------------------------------------------------
extern "C" void kernel_launch(void* const* d_in, const int* in_sizes, int n_in,
                              void* d_out, int out_size, void* d_ws, size_t ws_size,
                              hipStream_t stream) {
  const float* xs = (const float*)d_in[0];
  const float* wq = (const float*)d_in[1];
  const float* wk = (const float*)d_in[2];
  const float* wv = (const float*)d_in[3];
  const float* wo = (const float*)d_in[4];
  const float* ak = (const float*)d_in[5];
  float* out = (float*)d_out;

  // workspace layout (floats): Q,K,V,Z = BHS*64 each; P = BHS*272
  constexpr size_t SZ = (size_t)B_ * H_ * S_ * D_;   // 2,097,152
  float* ws = (float*)d_ws;
  float* Q = ws;
  float* K = Q + SZ;
  float* V = K + SZ;
  float* Z = V + SZ;
  float* P = Z + SZ;                                  // BHS*272 floats

  qkv_proj_kernel<<<dim3(S_ / 128, H_, B_), 256, 0, stream>>>(xs, wq, wk, wv, Q, K, V);
  qrel_kernel<<<dim3(S_ / 128, H_, B_), 256, 0, stream>>>(Q, ak, P);
  attn_kernel<<<dim3(S_ / 64, H_, B_), 128, 0, stream>>>(Q, K, V, P, Z);
  head_reduce_kernel<<<(B_ * S_ * D_) / 256, 256, 0, stream>>>(Z, wo, out);
}